// spConvnet_HSCDC_subM_41223096107608
// MI455X (gfx1250) — compile-verified
//
#include <hip/hip_runtime.h>
#include <math.h>

// ---------------------------------------------------------------------------
// Sparse 3D CNN forward for MI455X (gfx1250), wave32, f32 WMMA 16x16x4.
// ---------------------------------------------------------------------------

typedef float v8f __attribute__((ext_vector_type(8)));
typedef float v2f __attribute__((ext_vector_type(2)));

#define S3 120
#define NPTS 10000
#define NC3 195112            // 58^3
#define NT3 12195             // ceil(NC3/16)
#define NCP 24389             // 29^3
#define NC4 2744              // 14^3
#define NC5 1728              // 12^3

static __device__ __forceinline__ v8f wmma4(v2f a, v2f b, v8f c) {
  // D = A(16x4 f32) * B(4x16 f32) + C(16x16 f32)
  return __builtin_amdgcn_wmma_f32_16x16x4_f32(
      /*neg_a=*/false, a, /*neg_b=*/false, b,
      /*c_mod=*/(short)0, c, /*reuse_a=*/false, /*reuse_b=*/false);
}

static __device__ __forceinline__ unsigned ballot32(bool p) {
  return __builtin_amdgcn_ballot_w32(p);
}

static __device__ __forceinline__ int gidx(const int* __restrict__ g,
                                           int z, int y, int x) {
  if ((unsigned)z < (unsigned)S3 && (unsigned)y < (unsigned)S3 &&
      (unsigned)x < (unsigned)S3)
    return g[(z * S3 + y) * S3 + x];
  return 0;
}

// --------------------------------- utility ---------------------------------

__global__ void k_zero(float* __restrict__ p, int n) {
  int i = blockIdx.x * blockDim.x + threadIdx.x;
  if (i < n) p[i] = 0.0f;
}

__global__ void k_scatter(const int* __restrict__ inds, int* __restrict__ grid) {
  int p = blockIdx.x * blockDim.x + threadIdx.x;
  if (p >= NPTS) return;
  int z = inds[p * 4 + 1], y = inds[p * 4 + 2], x = inds[p * 4 + 3];
  atomicMax(&grid[(z * S3 + y) * S3 + x], p + 1);  // deterministic winner
}

// ------------------------- conv1: 7^3, 1->32, subM -------------------------
// GEMM: [N x 344] x [344 x 32], A gathered from point grid. 625 tiles x 1 wave.
// Tap decomposition staged in an LDS table: the hot loop is ds_load + bfe,
// no integer div/mod.

__global__ __launch_bounds__(32) void k_conv1(
    const int* __restrict__ grid, const float* __restrict__ feats,
    const int* __restrict__ inds, const float* __restrict__ w1,
    const float* __restrict__ b1, float* __restrict__ feat1) {
  __shared__ int ttab[344];
  const int lane = threadIdx.x;
  const int m = lane & 15;
  const int half = lane >> 4;
  const int tile = blockIdx.x;
  const int site = tile * 16 + m;  // 625*16 == 10000 exactly
  const int z = inds[site * 4 + 1];
  const int y = inds[site * 4 + 2];
  const int x = inds[site * 4 + 3];

  // build tap table once (div/mod out of the hot loop)
  for (int i = lane; i < 344; i += 32) {
    if (i < 343) {
      int dz = i / 49, r = i % 49, dy = r / 7, dx = r % 7;
      ttab[i] = (dz << 16) | (dy << 8) | dx;
    } else {
      ttab[i] = -1;  // zero-pad K to 344
    }
  }
  __syncthreads();

  v8f acc0 = {0, 0, 0, 0, 0, 0, 0, 0};
  v8f acc1 = {0, 0, 0, 0, 0, 0, 0, 0};

  for (int kc = 0; kc < 86; ++kc) {
    const int k0 = kc * 4 + half * 2;  // A layout: halves hold K pairs
    const int2 tt = *(const int2*)&ttab[k0];  // ds_load_b64
    v2f a;
#pragma unroll
    for (int j = 0; j < 2; ++j) {
      const int t = (j == 0) ? tt.x : tt.y;
      float v = 0.0f;
      if (t >= 0) {
        int pid = gidx(grid, z + (t >> 16) - 3, y + ((t >> 8) & 255) - 3,
                       x + (t & 255) - 3);
        if (pid) v = feats[pid - 1];
      }
      if (j == 0) a.x = v; else a.y = v;
    }
    v2f b0, b1v;
#pragma unroll
    for (int j = 0; j < 2; ++j) {
      int k = k0 + j;
      float u0 = 0.0f, u1 = 0.0f;
      if (k < 343) {
        u0 = w1[k * 32 + m];
        u1 = w1[k * 32 + 16 + m];
      }
      if (j == 0) { b0.x = u0; b1v.x = u1; } else { b0.y = u0; b1v.y = u1; }
    }
    acc0 = wmma4(a, b0, acc0);
    acc1 = wmma4(a, b1v, acc1);
  }

#pragma unroll
  for (int r = 0; r < 8; ++r) {
    const int row = r + half * 8;
    const int s = tile * 16 + row;
    feat1[s * 32 + m] = fmaxf(acc0[r] + b1[m], 0.0f);
    feat1[s * 32 + 16 + m] = fmaxf(acc1[r] + b1[16 + m], 0.0f);
  }
}

// ------------------------ conv2: 5^3, 32->40, subM --------------------------
// GEMM: [N x 4000] x [4000 x 40]; 3 accumulators share the gathered A row.
// Nested dz/dy/dx loops: no div/mod, coordinates by adds, tap counter runs
// in w-layout order.

__global__ __launch_bounds__(32) void k_conv2(
    const int* __restrict__ grid, const int* __restrict__ inds,
    const float* __restrict__ feat1, const float* __restrict__ w2,
    const float* __restrict__ b2, float* __restrict__ feat2) {
  const int lane = threadIdx.x;
  const int m = lane & 15;
  const int half = lane >> 4;
  const int tile = blockIdx.x;
  const int site = tile * 16 + m;
  const int z = inds[site * 4 + 1];
  const int y = inds[site * 4 + 2];
  const int x = inds[site * 4 + 3];

  v8f acc[3];
#pragma unroll
  for (int t = 0; t < 3; ++t) acc[t] = (v8f){0, 0, 0, 0, 0, 0, 0, 0};

  int tap = 0;
  for (int dz = -2; dz <= 2; ++dz)
    for (int dy = -2; dy <= 2; ++dy)
      for (int dx = -2; dx <= 2; ++dx, ++tap) {
        const int pid = gidx(grid, z + dz, y + dy, x + dx);
        const unsigned bal = ballot32(pid != 0);
        if ((bal & 0xFFFFu) == 0u) continue;  // wave-uniform tap skip
        const float* arow = pid ? &feat1[(pid - 1) * 32] : feat1;
        const int kbase = tap * 32;
        for (int c0 = 0; c0 < 32; c0 += 4) {
          const int ka = c0 + half * 2;
          const float2 av =
              pid ? *(const float2*)(arow + ka) : make_float2(0.f, 0.f);
          v2f a;
          a.x = av.x;
          a.y = av.y;
          const int kb = kbase + ka;
#pragma unroll
          for (int t = 0; t < 3; ++t) {
            const int n = t * 16 + m;
            v2f b;
            b.x = (n < 40) ? w2[kb * 40 + n] : 0.0f;
            b.y = (n < 40) ? w2[(kb + 1) * 40 + n] : 0.0f;
            acc[t] = wmma4(a, b, acc[t]);
          }
        }
      }

#pragma unroll
  for (int r = 0; r < 8; ++r) {
    const int row = r + half * 8;
    const int s = tile * 16 + row;
#pragma unroll
    for (int t = 0; t < 3; ++t) {
      const int n = t * 16 + m;
      if (n < 40) feat2[s * 40 + n] = fmaxf(acc[t][r] + b2[n], 0.0f);
    }
  }
}

// -------------------- dilated mask for conv3 (6^3, stride 2) ----------------

__global__ void k_mask3(const int* __restrict__ inds, int* __restrict__ mask3) {
  int p = blockIdx.x * blockDim.x + threadIdx.x;
  if (p >= NPTS) return;
  int z = inds[p * 4 + 1], y = inds[p * 4 + 2], x = inds[p * 4 + 3];
  int zl = (z > 5) ? ((z - 4) >> 1) : 0, zh = (z >> 1) < 57 ? (z >> 1) : 57;
  int yl = (y > 5) ? ((y - 4) >> 1) : 0, yh = (y >> 1) < 57 ? (y >> 1) : 57;
  int xl = (x > 5) ? ((x - 4) >> 1) : 0, xh = (x >> 1) < 57 ? (x >> 1) : 57;
  for (int oz = zl; oz <= zh; ++oz)
    for (int oy = yl; oy <= yh; ++oy)
      for (int ox = xl; ox <= xh; ++ox)
        mask3[(oz * 58 + oy) * 58 + ox] = 1;
}

// --------------------- conv3: 6^3 stride 2, 40->50, WMMA --------------------
// GEMM per cell tile: [16 cells x 8640] x [8640 x 64(=50 pad)], tap skipping.

__global__ __launch_bounds__(32) void k_conv3(
    const int* __restrict__ grid, const int* __restrict__ mask3,
    const float* __restrict__ feat2, const float* __restrict__ w3,
    const float* __restrict__ b3, float* __restrict__ x3) {
  const int lane = threadIdx.x;
  const int m = lane & 15;
  const int half = lane >> 4;
  const int tile = blockIdx.x;
  const int cell = tile * 16 + m;
  const bool valid = cell < NC3;
  int bz = 0, by = 0, bx = 0;
  if (valid) {
    int oz = cell / 3364;
    int r = cell % 3364;
    bz = 2 * oz;
    by = 2 * (r / 58);
    bx = 2 * (r % 58);
  }
  const int mk = valid ? mask3[cell] : 0;
  unsigned bal = ballot32(mk != 0);
  if ((bal & 0xFFFFu) == 0u) return;  // whole tile masked off (x3 pre-zeroed)

  v8f acc[4];
#pragma unroll
  for (int t = 0; t < 4; ++t) acc[t] = (v8f){0, 0, 0, 0, 0, 0, 0, 0};

  int tap = 0;
  for (int dz = 0; dz < 6; ++dz)
    for (int dy = 0; dy < 6; ++dy)
      for (int dx = 0; dx < 6; ++dx, ++tap) {
        const int pid = valid ? gidx(grid, bz + dz, by + dy, bx + dx) : 0;
        bal = ballot32(pid != 0);
        if ((bal & 0xFFFFu) == 0u) continue;  // empty tap across all 16 cells
        const float* arow = pid ? &feat2[(pid - 1) * 40] : feat2;
        const int kbase = tap * 40;
        for (int c0 = 0; c0 < 40; c0 += 4) {
          const int ka = c0 + half * 2;
          const float2 av =
              pid ? *(const float2*)(arow + ka) : make_float2(0.f, 0.f);
          v2f a;
          a.x = av.x;
          a.y = av.y;
          const int kb = kbase + ka;
#pragma unroll
          for (int t = 0; t < 4; ++t) {
            const int n = t * 16 + m;
            v2f b;
            b.x = (n < 50) ? w3[kb * 50 + n] : 0.0f;
            b.y = (n < 50) ? w3[(kb + 1) * 50 + n] : 0.0f;
            acc[t] = wmma4(a, b, acc[t]);
          }
        }
      }

#pragma unroll
  for (int r = 0; r < 8; ++r) {
    const int row = r + half * 8;
    const int c2 = tile * 16 + row;
    if (c2 < NC3 && mask3[c2]) {
#pragma unroll
      for (int t = 0; t < 4; ++t) {
        const int n = t * 16 + m;
        if (n < 50) x3[c2 * 50 + n] = fmaxf(acc[t][r] + b3[n], 0.0f);
      }
    }
  }
}

// ----------------- maxpool 58^3->29^3 (x3 and mask3 together) ---------------

__global__ void k_pool1(const float* __restrict__ x3,
                        const int* __restrict__ mask3, float* __restrict__ xp,
                        int* __restrict__ mp) {
  int tid = blockIdx.x * blockDim.x + threadIdx.x;
  if (tid < NCP * 50) {
    int cell = tid / 50, c = tid % 50;
    int oz = cell / 841, r = cell % 841, oy = r / 29, ox = r % 29;
    float v = -1e30f;
#pragma unroll
    for (int d = 0; d < 8; ++d) {
      int dz = d >> 2, dy = (d >> 1) & 1, dx = d & 1;
      int ic = ((2 * oz + dz) * 58 + (2 * oy + dy)) * 58 + (2 * ox + dx);
      v = fmaxf(v, x3[ic * 50 + c]);
    }
    xp[tid] = v;
  } else if (tid < NCP * 51) {
    int cell = tid - NCP * 50;
    int oz = cell / 841, r = cell % 841, oy = r / 29, ox = r % 29;
    int v = 0;
#pragma unroll
    for (int d = 0; d < 8; ++d) {
      int dz = d >> 2, dy = (d >> 1) & 1, dx = d & 1;
      v |= mask3[((2 * oz + dz) * 58 + (2 * oy + dy)) * 58 + (2 * ox + dx)];
    }
    mp[cell] = v;
  }
}

__global__ void k_mask4(const int* __restrict__ mp, int* __restrict__ mask4) {
  int tid = blockIdx.x * blockDim.x + threadIdx.x;
  if (tid >= NC4) return;
  int oz = tid / 196, oy = (tid / 14) % 14, ox = tid % 14;
  int v = 0;
  for (int dz = 0; dz < 3; ++dz)
    for (int dy = 0; dy < 3; ++dy)
      for (int dx = 0; dx < 3; ++dx)
        v |= mp[((2 * oz + dz) * 29 + (2 * oy + dy)) * 29 + (2 * ox + dx)];
  mask4[tid] = v ? 1 : 0;
}

// --------------------- conv4: 3^3 stride 2, 50->30 --------------------------

__global__ void k_conv4(const float* __restrict__ xp,
                        const int* __restrict__ mp,
                        const int* __restrict__ mask4,
                        const float* __restrict__ w4,
                        const float* __restrict__ b4, float* __restrict__ x4) {
  int tid = blockIdx.x * blockDim.x + threadIdx.x;
  if (tid >= NC4 * 30) return;
  int ocell = tid / 30, co = tid % 30;
  if (!mask4[ocell]) { x4[tid] = 0.0f; return; }
  int oz = ocell / 196, oy = (ocell / 14) % 14, ox = ocell % 14;
  float acc = b4[co];
  int d = 0;
  for (int dz = 0; dz < 3; ++dz)
    for (int dy = 0; dy < 3; ++dy)
      for (int dx = 0; dx < 3; ++dx, ++d) {
        int ic = ((2 * oz + dz) * 29 + (2 * oy + dy)) * 29 + (2 * ox + dx);
        if (!mp[ic]) continue;
        const float* row = &xp[ic * 50];
        const float* wr = &w4[d * 50 * 30 + co];
        for (int ci = 0; ci < 50; ++ci) acc += row[ci] * wr[ci * 30];
      }
  x4[tid] = fmaxf(acc, 0.0f);
}

__global__ void k_mask5(const int* __restrict__ mask4, int* __restrict__ mask5) {
  int tid = blockIdx.x * blockDim.x + threadIdx.x;
  if (tid >= NC5) return;
  int oz = tid / 144, oy = (tid / 12) % 12, ox = tid % 12;
  int v = 0;
  for (int dz = 0; dz < 3; ++dz)
    for (int dy = 0; dy < 3; ++dy)
      for (int dx = 0; dx < 3; ++dx)
        v |= mask4[((oz + dz) * 14 + (oy + dy)) * 14 + (ox + dx)];
  mask5[tid] = v ? 1 : 0;
}

// --------------------- conv5: 3^3 stride 1, 30->10 --------------------------

__global__ void k_conv5(const float* __restrict__ x4,
                        const int* __restrict__ mask4,
                        const int* __restrict__ mask5,
                        const float* __restrict__ w5,
                        const float* __restrict__ b5, float* __restrict__ x5) {
  int tid = blockIdx.x * blockDim.x + threadIdx.x;
  if (tid >= NC5 * 10) return;
  int ocell = tid / 10, co = tid % 10;
  if (!mask5[ocell]) { x5[tid] = 0.0f; return; }
  int oz = ocell / 144, oy = (ocell / 12) % 12, ox = ocell % 12;
  float acc = b5[co];
  int d = 0;
  for (int dz = 0; dz < 3; ++dz)
    for (int dy = 0; dy < 3; ++dy)
      for (int dx = 0; dx < 3; ++dx, ++d) {
        int ic = ((oz + dz) * 14 + (oy + dy)) * 14 + (ox + dx);
        if (!mask4[ic]) continue;
        const float* row = &x4[ic * 30];
        const float* wr = &w5[d * 30 * 10 + co];
        for (int ci = 0; ci < 30; ++ci) acc += row[ci] * wr[ci * 10];
      }
  x5[tid] = fmaxf(acc, 0.0f);
}

// --------------------------- FC head (one block) ----------------------------

__global__ __launch_bounds__(256) void k_head(
    const float* __restrict__ x5, const float* __restrict__ f1w,
    const float* __restrict__ f1b, const float* __restrict__ f21w,
    const float* __restrict__ f21b, const float* __restrict__ f31w,
    const float* __restrict__ f31b, const float* __restrict__ f41w,
    const float* __restrict__ f41b, const float* __restrict__ f22w,
    const float* __restrict__ f22b, const float* __restrict__ f32w,
    const float* __restrict__ f32b, const float* __restrict__ f42w,
    const float* __restrict__ f42b, float* __restrict__ out) {
  __shared__ float sxin[2160];
  __shared__ float sh[500];
  __shared__ float sa[200];
  __shared__ float sb[200];
  __shared__ float sy1[50];
  __shared__ float sy2[50];
  __shared__ float so[4];
  const int tid = threadIdx.x;

  // maxpool 12^3->6^3 + channel-major flatten (transpose (0,4,1,2,3))
  for (int i = tid; i < 2160; i += 256) {
    int c = i / 216, r = i % 216, z = r / 36, y = (r % 36) / 6, x = r % 6;
    float v = -1e30f;
#pragma unroll
    for (int d = 0; d < 8; ++d) {
      int dz = d >> 2, dy = (d >> 1) & 1, dx = d & 1;
      int ic = ((2 * z + dz) * 12 + (2 * y + dy)) * 12 + (2 * x + dx);
      v = fmaxf(v, x5[ic * 10 + c]);
    }
    sxin[i] = v;
  }
  __syncthreads();

  for (int j = tid; j < 500; j += 256) {
    float acc = f1b[j];
    const float* wr = &f1w[j * 2160];
    for (int k = 0; k < 2160; ++k) acc += wr[k] * sxin[k];
    sh[j] = fmaxf(acc, 0.0f);
  }
  __syncthreads();

  for (int j = tid; j < 400; j += 256) {
    if (j < 200) {
      float acc = f21b[j];
      const float* wr = &f21w[j * 500];
      for (int k = 0; k < 500; ++k) acc += wr[k] * sh[k];
      sa[j] = fmaxf(acc, 0.0f);
    } else {
      int jj = j - 200;
      float acc = f22b[jj];
      const float* wr = &f22w[jj * 500];
      for (int k = 0; k < 500; ++k) acc += wr[k] * sh[k];
      sb[jj] = fmaxf(acc, 0.0f);
    }
  }
  __syncthreads();

  if (tid < 100) {
    if (tid < 50) {
      float acc = f31b[tid];
      const float* wr = &f31w[tid * 200];
      for (int k = 0; k < 200; ++k) acc += wr[k] * sa[k];
      sy1[tid] = tanhf(acc);
    } else {
      int jj = tid - 50;
      float acc = f32b[jj];
      const float* wr = &f32w[jj * 200];
      for (int k = 0; k < 200; ++k) acc += wr[k] * sb[k];
      sy2[jj] = fmaxf(acc, 0.0f);
    }
  }
  __syncthreads();

  if (tid < 4) {
    if (tid < 3) {
      float acc = f41b[tid];
      const float* wr = &f41w[tid * 50];
      for (int k = 0; k < 50; ++k) acc += wr[k] * sy1[k];
      so[tid] = acc;
    } else {
      float acc = f42b[0];
      for (int k = 0; k < 50; ++k) acc += f42w[k] * sy2[k];
      so[3] = acc;
    }
  }
  __syncthreads();

  if (tid == 0) {
    float n = sqrtf(so[0] * so[0] + so[1] * so[1] + so[2] * so[2]);
    n = fmaxf(n, 1e-12f);
    out[0] = so[0] / n;
    out[1] = so[1] / n;
    out[2] = so[2] / n;
    float sp = so[3];
    out[3] = (sp > 30.0f) ? sp : log1pf(expf(sp));
  }
}

// ------------------------------- host driver --------------------------------

static inline size_t alignup(size_t v) { return (v + 255) & ~size_t(255); }

extern "C" void kernel_launch(void* const* d_in, const int* in_sizes, int n_in,
                              void* d_out, int out_size, void* d_ws,
                              size_t ws_size, hipStream_t stream) {
  const float* feats = (const float*)d_in[0];
  const int* inds = (const int*)d_in[1];
  const float* w1 = (const float*)d_in[3];
  const float* b1 = (const float*)d_in[4];
  const float* w2 = (const float*)d_in[5];
  const float* b2 = (const float*)d_in[6];
  const float* w3 = (const float*)d_in[7];
  const float* b3 = (const float*)d_in[8];
  const float* w4 = (const float*)d_in[9];
  const float* b4 = (const float*)d_in[10];
  const float* w5 = (const float*)d_in[11];
  const float* b5 = (const float*)d_in[12];
  const float* f1w = (const float*)d_in[13];
  const float* f1b = (const float*)d_in[14];
  const float* f21w = (const float*)d_in[15];
  const float* f21b = (const float*)d_in[16];
  const float* f31w = (const float*)d_in[17];
  const float* f31b = (const float*)d_in[18];
  const float* f41w = (const float*)d_in[19];
  const float* f41b = (const float*)d_in[20];
  const float* f22w = (const float*)d_in[21];
  const float* f22b = (const float*)d_in[22];
  const float* f32w = (const float*)d_in[23];
  const float* f32b = (const float*)d_in[24];
  const float* f42w = (const float*)d_in[25];
  const float* f42b = (const float*)d_in[26];

  // workspace bump allocator
  char* base = (char*)d_ws;
  size_t cur = 0;
  auto take = [&](size_t elems) {
    void* p = base + cur;
    cur = alignup(cur + elems * 4);
    return p;
  };
  int* grid = (int*)take(S3 * S3 * S3);     // point-id grid (120^3)
  int* mask3 = (int*)take(NC3);             // 58^3 dilated mask
  float* x3 = (float*)take(NC3 * 50);       // conv3 output (dense 58^3 x 50)
  float* feat1 = (float*)take(NPTS * 32);   // conv1 out at points
  float* feat2 = (float*)take(NPTS * 40);   // conv2 out at points
  float* xp = (float*)take(NCP * 50);       // pooled 29^3 x 50
  int* mp = (int*)take(NCP);                // pooled mask 29^3
  int* mask4 = (int*)take(NC4);
  float* x4 = (float*)take(NC4 * 30);
  int* mask5 = (int*)take(NC5);
  float* x5 = (float*)take(NC5 * 10);
  (void)ws_size;

  const int B = 256;
  // re-zero read-before-write regions every call (graph replay safe)
  k_zero<<<(S3 * S3 * S3 + B - 1) / B, B, 0, stream>>>((float*)grid,
                                                       S3 * S3 * S3);
  k_zero<<<(NC3 + B - 1) / B, B, 0, stream>>>((float*)mask3, NC3);
  k_zero<<<(NC3 * 50 + B - 1) / B, B, 0, stream>>>(x3, NC3 * 50);

  k_scatter<<<(NPTS + B - 1) / B, B, 0, stream>>>(inds, grid);

  k_conv1<<<NPTS / 16, 32, 0, stream>>>(grid, feats, inds, w1, b1, feat1);
  k_conv2<<<NPTS / 16, 32, 0, stream>>>(grid, inds, feat1, w2, b2, feat2);

  k_mask3<<<(NPTS + B - 1) / B, B, 0, stream>>>(inds, mask3);
  k_conv3<<<NT3, 32, 0, stream>>>(grid, mask3, feat2, w3, b3, x3);

  k_pool1<<<(NCP * 51 + B - 1) / B, B, 0, stream>>>(x3, mask3, xp, mp);
  k_mask4<<<(NC4 + B - 1) / B, B, 0, stream>>>(mp, mask4);
  k_conv4<<<(NC4 * 30 + B - 1) / B, B, 0, stream>>>(xp, mp, mask4, w4, b4, x4);
  k_mask5<<<(NC5 + B - 1) / B, B, 0, stream>>>(mask4, mask5);
  k_conv5<<<(NC5 * 10 + B - 1) / B, B, 0, stream>>>(x4, mask4, mask5, w5, b5,
                                                   x5);

  k_head<<<1, 256, 0, stream>>>(x5, f1w, f1b, f21w, f21b, f31w, f31b, f41w,
                                f41b, f22w, f22b, f32w, f32b, f42w, f42b,
                                (float*)d_out);
}